// GCKANLayer_51994874085900
// MI455X (gfx1250) — compile-verified
//
#include <hip/hip_runtime.h>
#include <hip/hip_bf16.h>

#define N_NODES 50000
#define N_EDGES 800000
#define IN_C    128
#define OUT_C   128
#define HIDDEN  16
#define N_COEF  7

typedef __attribute__((ext_vector_type(2))) float v2f;
typedef __attribute__((ext_vector_type(8))) float v8f;

// ---------------------------------------------------------------------------
// fp32 WMMA 16x16x4 (V_WMMA_F32_16X16X4_F32), exact fp32 math.
// ---------------------------------------------------------------------------
__device__ __forceinline__ v8f wmma4(v2f a, v2f b, v8f c) {
  return __builtin_amdgcn_wmma_f32_16x16x4_f32(false, a, false, b, (short)0, c,
                                               false, false);
}

__device__ __forceinline__ float silu_f(float v) {
  return v * (1.0f / (1.0f + __expf(-v)));
}

// Cox–de Boor, uniform grid g(i)=0.5*i-2.5 (GRID=4, order 3) -> 7 bases
__device__ __forceinline__ void bspline7(float v, float* out) {
  float b[10];
#pragma unroll
  for (int j = 0; j < 10; ++j) {
    float gj = 0.5f * (float)j - 2.5f;
    b[j] = (v >= gj && v < gj + 0.5f) ? 1.0f : 0.0f;
  }
#pragma unroll
  for (int p = 1; p <= 3; ++p) {
    float invd = 1.0f / (0.5f * (float)p);
#pragma unroll
    for (int j = 0; j < 10 - p; ++j) {
      float gj  = 0.5f * (float)j - 2.5f;
      float gjp = gj + 0.5f * (float)p;
      b[j] = (v - gj) * invd * b[j] + ((gjp + 0.5f) - v) * invd * b[j + 1];
    }
  }
#pragma unroll
  for (int c = 0; c < N_COEF; ++c) out[c] = b[c];
}

// ---------------------------------------------------------------------------
// Kernel 1: hlin = x @ W^T (50000x128 @ 128x128).
// Each wave: 4 row tiles x 1 col tile -> 4 independent WMMA chains sharing
// one B fragment (4x weight-load reuse, hides WMMA RAW hazard latency).
// ---------------------------------------------------------------------------
__global__ void k_gemm_xw(const float* __restrict__ x, const float* __restrict__ W,
                          float* __restrict__ hlin) {
  const int NT = N_NODES / 16;   // 3125 row tiles
  const int NG = (NT + 3) / 4;   // 782 groups of 4 row tiles
  int wave = (int)((blockIdx.x * blockDim.x + threadIdx.x) >> 5);
  int lane = (int)(threadIdx.x & 31);
  if (wave >= NG * (OUT_C / 16)) return;
  int grp = wave >> 3, ct = wave & 7;
  int o0 = ct * 16;
  int half = lane >> 4, l16 = lane & 15;

  int  rt[4];
  bool valid[4];
#pragma unroll
  for (int i = 0; i < 4; ++i) {
    int r = grp * 4 + i;
    valid[i] = (r < NT);
    rt[i] = valid[i] ? r : (NT - 1);   // clamp: duplicate compute, store skipped
  }

  v8f acc[4] = {};
  for (int k = 0; k < IN_C; k += 4) {
    int ka0 = k + (half ? 2 : 0);
    int ka1 = k + (half ? 3 : 1);
    v2f b;
    b.x = W[(o0 + l16) * IN_C + ka0];
    b.y = W[(o0 + l16) * IN_C + ka1];
#pragma unroll
    for (int i = 0; i < 4; ++i) {
      v2f a;
      a.x = x[(rt[i] * 16 + l16) * IN_C + ka0];
      a.y = x[(rt[i] * 16 + l16) * IN_C + ka1];
      acc[i] = wmma4(a, b, acc[i]);
    }
  }
#pragma unroll
  for (int i = 0; i < 4; ++i) {
    if (!valid[i]) continue;
#pragma unroll
    for (int j = 0; j < 8; ++j)
      hlin[(rt[i] * 16 + j + half * 8) * OUT_C + o0 + l16] = acc[i][j];
  }
}

// ---------------------------------------------------------------------------
// Degree / normalization kernels
// ---------------------------------------------------------------------------
__global__ void k_deg_init(float* __restrict__ deg) {
  int i = blockIdx.x * blockDim.x + threadIdx.x;
  if (i < N_NODES) deg[i] = 1.0f;  // self loop
}

__global__ void k_deg_acc(const int* __restrict__ ei, float* __restrict__ deg) {
  int e = blockIdx.x * blockDim.x + threadIdx.x;
  if (e < N_EDGES)
    __hip_atomic_fetch_add(&deg[ei[N_EDGES + e]], 1.0f, __ATOMIC_RELAXED,
                           __HIP_MEMORY_SCOPE_AGENT);
}

__global__ void k_deg_rsqrt(float* __restrict__ deg) {
  int i = blockIdx.x * blockDim.x + threadIdx.x;
  if (i < N_NODES) {
    float d = deg[i];
    deg[i] = (d > 0.0f) ? rsqrtf(d) : 0.0f;
  }
}

// h = hlin * dinv^2 (self loop) + bias
__global__ void k_self_bias(const float* __restrict__ hlin,
                            const float* __restrict__ dinv,
                            const float* __restrict__ bias,
                            float* __restrict__ h) {
  int t = blockIdx.x * blockDim.x + threadIdx.x;
  if (t < N_NODES * OUT_C) {
    int n = t >> 7, c = t & 127;
    float di = dinv[n];
    h[t] = hlin[t] * di * di + bias[c];
  }
}

// one wave per edge; lane handles 4 channels (float4 gather, 4 fp32 atomics).
// global_prefetch on the gathered row overlaps the dependent dinv loads.
__global__ void k_edge_scatter(const int* __restrict__ ei,
                               const float* __restrict__ hlin,
                               const float* __restrict__ dinv,
                               float* __restrict__ h) {
  long long t = (long long)blockIdx.x * blockDim.x + threadIdx.x;
  int e = (int)(t >> 5);
  int lane = (int)(t & 31);
  if (e >= N_EDGES) return;
  int src = ei[e], dst = ei[N_EDGES + e];
  const float4* h4 = (const float4*)hlin;
  const float4* p = &h4[src * 32 + lane];
  __builtin_prefetch(p, 0, 3);               // -> global_prefetch_b8
  float norm = dinv[src] * dinv[dst];
  float4 v = *p;
  float* base = &h[dst * OUT_C + lane * 4];
  __hip_atomic_fetch_add(base + 0, v.x * norm, __ATOMIC_RELAXED, __HIP_MEMORY_SCOPE_AGENT);
  __hip_atomic_fetch_add(base + 1, v.y * norm, __ATOMIC_RELAXED, __HIP_MEMORY_SCOPE_AGENT);
  __hip_atomic_fetch_add(base + 2, v.z * norm, __ATOMIC_RELAXED, __HIP_MEMORY_SCOPE_AGENT);
  __hip_atomic_fetch_add(base + 3, v.w * norm, __ATOMIC_RELAXED, __HIP_MEMORY_SCOPE_AGENT);
}

// ---------------------------------------------------------------------------
// Fold scaler into spline weights; K-layout: k = f*8 + j, j==0 -> base weight,
// j=1..7 -> spline_w[o][f][j-1]*scaler[o][f].  Bmat is (K rows, O cols).
// ---------------------------------------------------------------------------
__global__ void k_prep_w0(const float* __restrict__ bw, const float* __restrict__ sw,
                          const float* __restrict__ sc, float* __restrict__ Bm) {
  int t = blockIdx.x * blockDim.x + threadIdx.x;  // k*16 + o
  if (t >= 8 * OUT_C * HIDDEN) return;
  int k = t >> 4, o = t & 15;
  int f = k >> 3, j = k & 7;
  float w = (j == 0) ? bw[o * OUT_C + f]
                     : sw[(o * OUT_C + f) * N_COEF + (j - 1)] * sc[o * OUT_C + f];
  Bm[k * HIDDEN + o] = w;
}

__global__ void k_prep_w1(const float* __restrict__ bw, const float* __restrict__ sw,
                          const float* __restrict__ sc, float* __restrict__ Bm) {
  int t = blockIdx.x * blockDim.x + threadIdx.x;  // k*128 + o
  if (t >= 8 * HIDDEN * OUT_C) return;
  int k = t >> 7, o = t & 127;
  int f = k >> 3, j = k & 7;
  float w = (j == 0) ? bw[o * HIDDEN + f]
                     : sw[(o * HIDDEN + f) * N_COEF + (j - 1)] * sc[o * HIDDEN + f];
  Bm[k * OUT_C + o] = w;
}

// ---------------------------------------------------------------------------
// KAN layer 0: per 16-node tile, K=1024 in 4 chunks of 256 (32 features each).
// feat LDS stride 258 floats -> banks advance by 2 per row (conflict-free A).
// Two interleaved accumulators to break the WMMA RAW chain.
// ---------------------------------------------------------------------------
#define LDS0_STRIDE 258
__global__ void __launch_bounds__(32) k_kan0(const float* __restrict__ h,
                                             const float* __restrict__ Bm,
                                             float* __restrict__ g) {
  __shared__ float feat[16 * LDS0_STRIDE];
  int n0 = blockIdx.x * 16;
  int lane = threadIdx.x;
  int half = lane >> 4, l16 = lane & 15;
  v8f accA = {}, accB = {};
  for (int ch = 0; ch < 4; ++ch) {
    int f0 = ch * 32;
    for (int idx = lane; idx < 16 * 32; idx += 32) {
      int row = idx & 15, fl = idx >> 4;
      float v = h[(n0 + row) * OUT_C + f0 + fl];
      float* d = &feat[row * LDS0_STRIDE + fl * 8];
      d[0] = silu_f(v);
      float bs[N_COEF];
      bspline7(v, bs);
#pragma unroll
      for (int c = 0; c < N_COEF; ++c) d[1 + c] = bs[c];
    }
    __syncthreads();
    for (int k = 0; k < 256; k += 8) {
#pragma unroll
      for (int s = 0; s < 2; ++s) {
        int ka0 = k + s * 4 + (half ? 2 : 0);
        int ka1 = k + s * 4 + (half ? 3 : 1);
        v2f a, b;
        a.x = feat[l16 * LDS0_STRIDE + ka0];
        a.y = feat[l16 * LDS0_STRIDE + ka1];
        b.x = Bm[(ch * 256 + ka0) * HIDDEN + l16];
        b.y = Bm[(ch * 256 + ka1) * HIDDEN + l16];
        if (s == 0) accA = wmma4(a, b, accA);
        else        accB = wmma4(a, b, accB);
      }
    }
    __syncthreads();
  }
  v8f acc = accA + accB;
#pragma unroll
  for (int i = 0; i < 8; ++i)
    g[(n0 + i + half * 8) * HIDDEN + l16] = acc[i];
}

// ---------------------------------------------------------------------------
// KAN layer 1: K=128 (16 feats * 8), N=128 (8 column tiles), per 16-node tile.
// ---------------------------------------------------------------------------
#define LDS1_STRIDE 130
__global__ void __launch_bounds__(32) k_kan1(const float* __restrict__ g,
                                             const float* __restrict__ Bm,
                                             float* __restrict__ out) {
  __shared__ float feat[16 * LDS1_STRIDE];
  int n0 = blockIdx.x * 16;
  int lane = threadIdx.x;
  int half = lane >> 4, l16 = lane & 15;
  for (int idx = lane; idx < 16 * HIDDEN; idx += 32) {
    int row = idx & 15, fl = idx >> 4;
    float v = g[(n0 + row) * HIDDEN + fl];
    float* d = &feat[row * LDS1_STRIDE + fl * 8];
    d[0] = silu_f(v);
    float bs[N_COEF];
    bspline7(v, bs);
#pragma unroll
    for (int c = 0; c < N_COEF; ++c) d[1 + c] = bs[c];
  }
  __syncthreads();
  for (int ct = 0; ct < 8; ++ct) {
    int o0 = ct * 16;
    v8f accA = {}, accB = {};
    for (int k = 0; k < 128; k += 8) {
#pragma unroll
      for (int s = 0; s < 2; ++s) {
        int ka0 = k + s * 4 + (half ? 2 : 0);
        int ka1 = k + s * 4 + (half ? 3 : 1);
        v2f a, b;
        a.x = feat[l16 * LDS1_STRIDE + ka0];
        a.y = feat[l16 * LDS1_STRIDE + ka1];
        b.x = Bm[ka0 * OUT_C + o0 + l16];
        b.y = Bm[ka1 * OUT_C + o0 + l16];
        if (s == 0) accA = wmma4(a, b, accA);
        else        accB = wmma4(a, b, accB);
      }
    }
    v8f acc = accA + accB;
#pragma unroll
    for (int i = 0; i < 8; ++i)
      out[(n0 + i + half * 8) * OUT_C + o0 + l16] = acc[i];
  }
}

// ---------------------------------------------------------------------------
extern "C" void kernel_launch(void* const* d_in, const int* in_sizes, int n_in,
                              void* d_out, int out_size, void* d_ws, size_t ws_size,
                              hipStream_t stream) {
  (void)in_sizes; (void)n_in; (void)out_size; (void)ws_size;
  const float* x       = (const float*)d_in[0];
  const int*   ei      = (const int*)d_in[1];
  const float* gcn_w   = (const float*)d_in[2];
  const float* gcn_b   = (const float*)d_in[3];
  const float* base_w0 = (const float*)d_in[4];
  const float* spl_w0  = (const float*)d_in[5];
  const float* scal0   = (const float*)d_in[6];
  const float* base_w1 = (const float*)d_in[7];
  const float* spl_w1  = (const float*)d_in[8];
  const float* scal1   = (const float*)d_in[9];
  float* out = (float*)d_out;

  float* ws    = (float*)d_ws;
  float* hlin  = ws;                               // N*128
  float* h     = hlin + (size_t)N_NODES * OUT_C;   // N*128
  float* deg   = h + (size_t)N_NODES * OUT_C;      // N  (becomes dinv)
  float* g     = deg + N_NODES;                    // N*16
  float* Bm0   = g + (size_t)N_NODES * HIDDEN;     // 1024*16
  float* Bm1   = Bm0 + 8 * OUT_C * HIDDEN;         // 128*128

  const int NT = N_NODES / 16;
  const int NG = (NT + 3) / 4;

  // 1) dense GEMM via fp32 WMMA (4 row tiles / wave, shared B fragments)
  k_gemm_xw<<<NG, 256, 0, stream>>>(x, gcn_w, hlin);
  // 2) degrees + normalization
  k_deg_init<<<(N_NODES + 255) / 256, 256, 0, stream>>>(deg);
  k_deg_acc<<<(N_EDGES + 255) / 256, 256, 0, stream>>>(ei, deg);
  k_deg_rsqrt<<<(N_NODES + 255) / 256, 256, 0, stream>>>(deg);
  k_self_bias<<<(N_NODES * OUT_C + 255) / 256, 256, 0, stream>>>(hlin, deg, gcn_b, h);
  k_edge_scatter<<<((long long)N_EDGES * 32 + 255) / 256, 256, 0, stream>>>(ei, hlin, deg, h);
  // 3) fold weights
  k_prep_w0<<<(8 * OUT_C * HIDDEN + 255) / 256, 256, 0, stream>>>(base_w0, spl_w0, scal0, Bm0);
  k_prep_w1<<<(8 * HIDDEN * OUT_C + 255) / 256, 256, 0, stream>>>(base_w1, spl_w1, scal1, Bm1);
  // 4) KAN layers via fp32 WMMA
  k_kan0<<<N_NODES / 16, 32, 0, stream>>>(h, Bm0, g);
  k_kan1<<<N_NODES / 16, 32, 0, stream>>>(g, Bm1, out);
}